// SGRUCell_32023276159533
// MI455X (gfx1250) — compile-verified
//
#include <hip/hip_runtime.h>
#include <math.h>

#define H_   256
#define I_   128
#define R_   32
#define B_   16
#define T_   48
#define H3   768
#define LN_EPS   1e-5f
#define CLIP_VAL 1.0f

typedef __attribute__((ext_vector_type(2))) float v2f;
typedef __attribute__((ext_vector_type(8))) float v8f;

__device__ __forceinline__ float sigmoidf_(float x) { return 1.0f / (1.0f + expf(-x)); }

// ---------------------------------------------------------------------------
// Kernel A1: Wx_raw = x @ W_x2h^T + b_x2h    (M=768 rows(T*B), N=768, K=128)
// One wave per 16x16 tile; fp32 WMMA 16x16x4, K-loop of 32.
// ---------------------------------------------------------------------------
__global__ __launch_bounds__(256)
void k_gemm_x(const float* __restrict__ X, const float* __restrict__ W,
              const float* __restrict__ bias, float* __restrict__ out)
{
    const int lane = threadIdx.x & 31;
    const int wave = threadIdx.x >> 5;
    const int wid  = blockIdx.x * 8 + wave;         // 0..2303
    const int tm = wid / 48, tn = wid % 48;
    const int row  = tm * 16 + (lane & 15);
    const int col  = tn * 16 + (lane & 15);
    const int koff = (lane >> 4) * 2;
    const int hi   = (lane >> 4) & 1;

    v8f c = {};
    for (int k = 0; k < I_; k += 4) {
        v2f a, b;
        a.x = X[row * I_ + k + koff];
        a.y = X[row * I_ + k + koff + 1];
        b.x = W[col * I_ + k + koff];
        b.y = W[col * I_ + k + koff + 1];
        c = __builtin_amdgcn_wmma_f32_16x16x4_f32(false, a, false, b, (short)0, c, false, false);
    }
    const float bn = bias[col];
    for (int v = 0; v < 8; ++v) {
        const int m = tm * 16 + v + 8 * hi;
        out[m * H3 + col] = c[v] + bn;
    }
}

// ---------------------------------------------------------------------------
// Kernel A2: in-place LayerNorm over rows of length 768 (ln_x)
// ---------------------------------------------------------------------------
__global__ __launch_bounds__(256)
void k_ln_x(float* __restrict__ y, const float* __restrict__ g, const float* __restrict__ bb)
{
    __shared__ float red[256];
    const int row = blockIdx.x;
    const int tid = threadIdx.x;
    float* yr = y + (size_t)row * H3;
    float v0 = yr[tid], v1 = yr[tid + 256], v2 = yr[tid + 512];
    red[tid] = v0 + v1 + v2;
    __syncthreads();
    for (int o = 128; o; o >>= 1) { if (tid < o) red[tid] += red[tid + o]; __syncthreads(); }
    const float mu = red[0] / 768.0f;
    __syncthreads();
    const float d0 = v0 - mu, d1 = v1 - mu, d2 = v2 - mu;
    red[tid] = d0 * d0 + d1 * d1 + d2 * d2;
    __syncthreads();
    for (int o = 128; o; o >>= 1) { if (tid < o) red[tid] += red[tid + o]; __syncthreads(); }
    const float rs = rsqrtf(red[0] / 768.0f + LN_EPS);
    yr[tid]       = d0 * rs * g[tid]       + bb[tid];
    yr[tid + 256] = d1 * rs * g[tid + 256] + bb[tid + 256];
    yr[tid + 512] = d2 * rs * g[tid + 512] + bb[tid + 512];
}

// ---------------------------------------------------------------------------
// Kernel B: per-step gates (single workgroup, 512 threads = 16 waves)
//   Wh = h@W_h2h^T + b (WMMA), += plastic, LN, pre = Wx+Wh, gates, h/v/te,
//   mod, s/m + per-step constants. Writes state + outputs.
// ---------------------------------------------------------------------------
__global__ __launch_bounds__(512)
void k_step_gates(int t,
    const float* __restrict__ wx,
    float* __restrict__ h_state, float* __restrict__ v_state, float* __restrict__ te_state,
    float* __restrict__ sm,
    const float* __restrict__ dU,
    const float* __restrict__ Wh, const float* __restrict__ bh,
    const float* __restrict__ lng, const float* __restrict__ lnb,
    const float* __restrict__ Wmod, const float* __restrict__ bmod,
    const float* __restrict__ Wm2h, const float* __restrict__ bm2h,
    const float* __restrict__ alpha_p, const float* __restrict__ tau_p,
    float* __restrict__ out_v, float* __restrict__ out_h, float* __restrict__ out_te,
    float* __restrict__ out_outs, float* __restrict__ out_mods)
{
    __shared__ float pre[B_][H3];     // 48 KB
    __shared__ float modl[B_][R_];    // 2 KB
    const int tid  = threadIdx.x;
    const int lane = tid & 31;
    const int wave = tid >> 5;        // 0..15
    const float sp_a = log1pf(expf(alpha_p[0]));

    // 1) Wh = h @ W_h2h^T + b_h2h via fp32 WMMA; wave handles N-tiles wave,+16,+32
    {
        const int rowb = lane & 15;           // batch
        const int koff = (lane >> 4) * 2;
        const int hi   = (lane >> 4) & 1;
        for (int tt = wave; tt < 48; tt += 16) {
            const int col = tt * 16 + (lane & 15);
            v8f c = {};
            for (int k = 0; k < H_; k += 4) {
                v2f a, b;
                a.x = h_state[rowb * H_ + k + koff];
                a.y = h_state[rowb * H_ + k + koff + 1];
                b.x = Wh[col * H_ + k + koff];
                b.y = Wh[col * H_ + k + koff + 1];
                c = __builtin_amdgcn_wmma_f32_16x16x4_f32(false, a, false, b, (short)0, c, false, false);
            }
            const float bn = bh[col];
            for (int v = 0; v < 8; ++v)
                pre[v + 8 * hi][col] = c[v] + bn;
        }
    }
    __syncthreads();

    // 2) plastic: pre[b][512+i] += sp_a * dot(dU[b][i][:], h[b][:])  (float4)
    for (int q = 0; q < 8; ++q) {
        const int e = tid + q * 512;
        const int b = e >> 8, i = e & 255;
        const float4* du4 = (const float4*)(dU + ((size_t)b * H_ + i) * H_);
        const float4* hh4 = (const float4*)(h_state + b * H_);
        float acc = 0.f;
        for (int j = 0; j < H_ / 4; ++j) {
            const float4 d = du4[j], h4 = hh4[j];
            acc += d.x * h4.x + d.y * h4.y + d.z * h4.z + d.w * h4.w;
        }
        pre[b][512 + i] += sp_a * acc;
    }
    __syncthreads();

    // 3) ln_h per row + add Wx: wave w owns row b=w (768 elems, 24/lane, shfl reduce)
    {
        const int b = wave;
        float vals[24];
        float s = 0.f;
        for (int q = 0; q < 24; ++q) { vals[q] = pre[b][lane + q * 32]; s += vals[q]; }
        for (int o = 16; o; o >>= 1) s += __shfl_xor(s, o, 32);
        const float mu = s / 768.0f;
        float vs = 0.f;
        for (int q = 0; q < 24; ++q) { const float d = vals[q] - mu; vs += d * d; }
        for (int o = 16; o; o >>= 1) vs += __shfl_xor(vs, o, 32);
        const float rs = rsqrtf(vs / 768.0f + LN_EPS);
        const float* wxr = wx + ((size_t)t * B_ + b) * H3;
        for (int q = 0; q < 24; ++q) {
            const int n = lane + q * 32;
            pre[b][n] = wxr[n] + (vals[q] - mu) * rs * lng[n] + lnb[n];
        }
    }
    __syncthreads();

    // 4) gates + state update (elementwise; stash h_new in pre[b][512+n])
    for (int q = 0; q < 8; ++q) {
        const int e = tid + q * 512;
        const int b = e >> 8, n = e & 255;
        const float z  = sigmoidf_(pre[b][n]);
        const float r  = sigmoidf_(pre[b][256 + n]);
        const float dv = pre[b][512 + n];
        const float vold  = v_state[b * H_ + n];
        const float hold  = h_state[b * H_ + n];
        const float teold = te_state[b * H_ + n];
        const float vnew  = (1.f - z) * vold + z * dv;
        const float hnew  = fmaxf(vnew, 0.f);
        const float tenew = (1.f - r) * teold + r * hold;
        v_state[b * H_ + n]  = vnew;
        h_state[b * H_ + n]  = hnew;
        te_state[b * H_ + n] = tenew;
        out_v[b * H_ + n]  = vnew;
        out_h[b * H_ + n]  = hnew;
        out_te[b * H_ + n] = tenew;
        out_outs[(size_t)t * B_ * H_ + b * H_ + n] = hnew;
        pre[b][512 + n] = hnew;
    }
    __syncthreads();

    // 5) mod = relu(h_new @ W_h2mod^T + b): one thread per (b, r)
    {
        const int b = tid >> 5, r = tid & 31;
        float acc = bmod[r];
        const float* wr = Wmod + r * H_;
        for (int j = 0; j < H_; ++j) acc += wr[j] * pre[b][512 + j];
        acc = fmaxf(acc, 0.f);
        modl[b][r] = acc;
        out_mods[(size_t)t * B_ * R_ + b * R_ + r] = acc;
    }
    __syncthreads();

    // 6) sm: s = sigmoid(mod@W0 + b0), m = mod@W1 + b1  (per batch)
    //    + per-step scalar constants for kernel C (slots 32..34)
    if (tid < 32) {
        const int b = tid >> 1, which = tid & 1;
        float acc = bm2h[which];
        for (int r = 0; r < R_; ++r) acc += Wm2h[which * R_ + r] * modl[b][r];
        sm[b * 2 + which] = (which == 0) ? sigmoidf_(acc) : acc;
    } else if (tid == 32) {
        sm[32] = sp_a;
        sm[33] = 1.0f / sp_a;
        sm[34] = sigmoidf_(tau_p[0]);
    }
}

// ---------------------------------------------------------------------------
// Kernel C: outer / trace_E / dU update with clip. float4 per thread.
// Block = 256 threads, handles 4 consecutive i-rows of one batch (1024 elems).
// Grid = B * H/4 = 1024 blocks. dU, tE updated in place in d_out (L2-resident).
// ---------------------------------------------------------------------------
__global__ __launch_bounds__(256)
void k_plastic_update(const float* __restrict__ h_state, const float* __restrict__ te_state,
                      const float* __restrict__ sm, const float* __restrict__ Wh,
                      float* __restrict__ dU, float* __restrict__ tE)
{
    const int bi = blockIdx.x;                  // 0..1023
    const int b = bi >> 6;                      // batch
    const int i = (bi & 63) * 4 + (threadIdx.x >> 6);   // row
    const int jv = threadIdx.x & 63;            // float4 column index (j = jv*4)

    const float inv_spa = sm[33];
    const float tau     = sm[34];
    const float s = sm[b * 2], m = sm[b * 2 + 1];
    const float om_s = 1.f - s, om_tau = 1.f - tau, tm = tau * m;

    const float hi_ = h_state[b * H_ + i], ti_ = te_state[b * H_ + i];
    const float4 hj4 = ((const float4*)(h_state + b * H_))[jv];
    const float4 tj4 = ((const float4*)(te_state + b * H_))[jv];

    const size_t v4 = ((size_t)b * H_ + i) * (H_ / 4) + jv;
    float4 du4 = ((float4*)dU)[v4];
    float4 te4 = ((float4*)tE)[v4];
    const float4 wv4 = ((const float4*)(Wh + (size_t)(2 * H_ + i) * H_))[jv];

    float*       du = (float*)&du4;
    float*       te = (float*)&te4;
    const float* hj = (const float*)&hj4;
    const float* tj = (const float*)&tj4;
    const float* wv = (const float*)&wv4;

#pragma unroll
    for (int q = 0; q < 4; ++q) {
        const float outer = hi_ * tj[q] - ti_ * hj[q];
        const float tE_new = om_s * te[q] + s * outer;
        float d = om_tau * du[q] + tm * tE_new;
        const float upper =  fmaxf(CLIP_VAL - wv[q], 0.f) * inv_spa;
        const float lower = -fmaxf(CLIP_VAL + wv[q], 0.f) * inv_spa;
        d = fminf(d, upper);
        d = fmaxf(d, lower);
        te[q] = tE_new;
        du[q] = d;
    }
    ((float4*)tE)[v4] = te4;
    ((float4*)dU)[v4] = du4;
}

// ---------------------------------------------------------------------------
extern "C" void kernel_launch(void* const* d_in, const int* in_sizes, int n_in,
                              void* d_out, int out_size, void* d_ws, size_t ws_size,
                              hipStream_t stream)
{
    const float* x      = (const float*)d_in[0];
    const float* h0     = (const float*)d_in[1];
    const float* v0     = (const float*)d_in[2];
    const float* dU0    = (const float*)d_in[3];
    const float* te0    = (const float*)d_in[4];
    const float* tE0    = (const float*)d_in[5];
    const float* Wx2h   = (const float*)d_in[6];
    const float* bx2h   = (const float*)d_in[7];
    const float* Wh2h   = (const float*)d_in[8];
    const float* bh2h   = (const float*)d_in[9];
    const float* lnxg   = (const float*)d_in[10];
    const float* lnxb   = (const float*)d_in[11];
    const float* lnhg   = (const float*)d_in[12];
    const float* lnhb   = (const float*)d_in[13];
    const float* Wh2mod = (const float*)d_in[14];
    const float* bh2mod = (const float*)d_in[15];
    const float* Wm2h   = (const float*)d_in[16];
    const float* bm2h   = (const float*)d_in[17];
    const float* alpha  = (const float*)d_in[18];
    const float* tauU   = (const float*)d_in[19];

    float* out = (float*)d_out;
    float* ws  = (float*)d_ws;

    // d_out layout (floats): v[0], h[4096], dU[8192], trace_e[1056768],
    //                        trace_E[1060864], outs[2109440], mods[2306048]
    float* dU = out + 8192;
    float* tE = out + 1060864;

    // d_ws layout (floats)
    float* wx  = ws;               // 768*768
    float* hs  = ws + 589824;      // 16*256
    float* vs  = hs + 4096;
    float* tes = vs + 4096;
    float* sm  = tes + 4096;       // 16*2 + 3 consts

    // re-init all state from inputs every call (deterministic)
    hipMemcpyAsync(hs,  h0,  (size_t)4096 * 4,    hipMemcpyDeviceToDevice, stream);
    hipMemcpyAsync(vs,  v0,  (size_t)4096 * 4,    hipMemcpyDeviceToDevice, stream);
    hipMemcpyAsync(tes, te0, (size_t)4096 * 4,    hipMemcpyDeviceToDevice, stream);
    hipMemcpyAsync(dU,  dU0, (size_t)1048576 * 4, hipMemcpyDeviceToDevice, stream);
    hipMemcpyAsync(tE,  tE0, (size_t)1048576 * 4, hipMemcpyDeviceToDevice, stream);

    // hoisted input projection: Wx_all = LN(x @ W_x2h^T + b)
    k_gemm_x<<<288, 256, 0, stream>>>(x, Wx2h, bx2h, wx);
    k_ln_x<<<768, 256, 0, stream>>>(wx, lnxg, lnxb);

    for (int t = 0; t < T_; ++t) {
        k_step_gates<<<1, 512, 0, stream>>>(t, wx, hs, vs, tes, sm, dU,
            Wh2h, bh2h, lnhg, lnhb, Wh2mod, bh2mod, Wm2h, bm2h, alpha, tauU,
            out + 0, out + 4096, out + 1056768, out + 2109440, out + 2306048);
        k_plastic_update<<<1024, 256, 0, stream>>>(hs, tes, sm, Wh2h, dU, tE);
    }
}